// ClassicalHybridClassifier_65481071402207
// MI455X (gfx1250) — compile-verified
//
#include <hip/hip_runtime.h>
#include <hip/hip_bf16.h>
#include <math.h>

typedef __attribute__((ext_vector_type(2))) float v2f;
typedef __attribute__((ext_vector_type(8))) float v8f;

#define BATCH   10
#define C1_CIN  3
#define C1_COUT 6
#define IN_H    250
#define IN_W    250
#define H1      124
#define W1      124
#define P1H     123
#define P1W     123
#define C2_CIN  6
#define C2_COUT 15
#define H2      62
#define W2      62
#define P2H     61
#define P2W     61
#define FC1_K   55815   // 15*61*61
#define FC1_KF  55812   // FC1_K & ~3 (full 4-wide steps)
#define FC1_N   120
#define FC2_N   84
#define NTRAIN  8192

// fc1 WMMA tiling: 8 N-tiles of 16 (120 valid cols), 128 K-chunks
#define FC1_NT    8
#define FC1_KT    128
#define FC1_STEPS 110   // ceil(ceil(55815/4)/128); 128*110*4 = 56320 >= 55815

// ---------------- conv1 + relu: (10,3,250,250) -> (10,6,124,124) ----------------
__global__ __launch_bounds__(256)
void conv1_kernel(const float* __restrict__ x, const float* __restrict__ w,
                  const float* __restrict__ b, float* __restrict__ out) {
    int idx = blockIdx.x * blockDim.x + threadIdx.x;
    const int total = BATCH * C1_COUT * H1 * W1;
    if (idx >= total) return;
    int xo = idx % W1;
    int yo = (idx / W1) % H1;
    int co = (idx / (W1 * H1)) % C1_COUT;
    int bb = idx / (W1 * H1 * C1_COUT);
    float acc = b[co];
    for (int ci = 0; ci < C1_CIN; ++ci) {
        const float* xp = x + (size_t)(bb * C1_CIN + ci) * IN_H * IN_W;
        const float* wp = w + (co * C1_CIN + ci) * 25;
        #pragma unroll
        for (int kh = 0; kh < 5; ++kh) {
            int ih = 2 * yo - 1 + kh;
            if (ih < 0 || ih >= IN_H) continue;
            #pragma unroll
            for (int kw = 0; kw < 5; ++kw) {
                int iw = 2 * xo - 1 + kw;
                if (iw < 0 || iw >= IN_W) continue;
                acc += xp[ih * IN_W + iw] * wp[kh * 5 + kw];
            }
        }
    }
    out[idx] = fmaxf(acc, 0.0f);
}

// ---------------- generic 2x2 stride-1 maxpool ----------------
__global__ __launch_bounds__(256)
void maxpool_kernel(const float* __restrict__ in, float* __restrict__ out,
                    int C, int Hin, int Win, int Hout, int Wout, int total) {
    int idx = blockIdx.x * blockDim.x + threadIdx.x;
    if (idx >= total) return;
    int xo = idx % Wout;
    int yo = (idx / Wout) % Hout;
    int c  = (idx / (Wout * Hout)) % C;
    int bb = idx / (Wout * Hout * C);
    const float* p = in + (((size_t)(bb * C + c) * Hin + yo) * Win + xo);
    float m = fmaxf(fmaxf(p[0], p[1]), fmaxf(p[Win], p[Win + 1]));
    out[idx] = m;
}

// ---------------- conv2 + relu: (10,6,123,123) -> (10,15,62,62) ----------------
__global__ __launch_bounds__(256)
void conv2_kernel(const float* __restrict__ x, const float* __restrict__ w,
                  const float* __restrict__ b, float* __restrict__ out) {
    int idx = blockIdx.x * blockDim.x + threadIdx.x;
    const int total = BATCH * C2_COUT * H2 * W2;
    if (idx >= total) return;
    int xo = idx % W2;
    int yo = (idx / W2) % H2;
    int co = (idx / (W2 * H2)) % C2_COUT;
    int bb = idx / (W2 * H2 * C2_COUT);
    float acc = b[co];
    for (int ci = 0; ci < C2_CIN; ++ci) {
        const float* xp = x + (size_t)(bb * C2_CIN + ci) * P1H * P1W;
        const float* wp = w + (co * C2_CIN + ci) * 9;
        #pragma unroll
        for (int kh = 0; kh < 3; ++kh) {
            int ih = 2 * yo - 1 + kh;
            if (ih < 0 || ih >= P1H) continue;
            #pragma unroll
            for (int kw = 0; kw < 3; ++kw) {
                int iw = 2 * xo - 1 + kw;
                if (iw < 0 || iw >= P1W) continue;
                acc += xp[ih * P1W + iw] * wp[kh * 3 + kw];
            }
        }
    }
    out[idx] = fmaxf(acc, 0.0f);
}

// ---------------- fc1 GEMM via V_WMMA_F32_16X16X4_F32 ----------------
// C(16x16 per N-tile) = A(16x4) * B(4x16) accumulated over a K chunk.
// A rows = batch (rows >= 10 clamped -> garbage, discarded by reduce).
// B[k][n] = fc1_w[n][k] (cols >= 120 clamped -> garbage, discarded).
// Straight-line inner loop: 4 unpredicated b32 loads + 1 wmma per K step.
__global__ __launch_bounds__(32)
void fc1_wmma_kernel(const float* __restrict__ act,   // (10, 55815)
                     const float* __restrict__ w,     // (120, 55815)
                     float* __restrict__ part) {
    const int nt   = blockIdx.x;        // 0..7
    const int kt   = blockIdx.y;        // 0..127
    const int lane = threadIdx.x;       // 0..31  (wave32)
    const int half = lane >> 4;         // ISA: lanes 16-31 hold K=2,3
    const int lm   = lane & 15;
    const int m_eff = (lm < BATCH) ? lm : (BATCH - 1);        // clamp: rows >=10 discarded
    const int n     = nt * 16 + lm;
    const int n_eff = (n < FC1_N) ? n : (FC1_N - 1);          // clamp: cols >=120 discarded

    const float* ap = act + (size_t)m_eff * FC1_K + 2 * half; // A[m][k + 2*half + {0,1}]
    const float* bp = w   + (size_t)n_eff * FC1_K + 2 * half; // B[k][n] = w[n][k]

    v8f c = {};
    const int kbeg  = kt * (FC1_STEPS * 4);
    const int kend  = (kbeg + FC1_STEPS * 4 < FC1_K) ? (kbeg + FC1_STEPS * 4) : FC1_K;
    const int kstop = (kend < FC1_KF) ? kend : FC1_KF;        // unchecked region

    int k = kbeg;
    #pragma unroll 2
    for (; k + 4 <= kstop; k += 4) {
        v2f a, b;
        a.x = ap[k];
        a.y = ap[k + 1];
        b.x = bp[k];
        b.y = bp[k + 1];
        c = __builtin_amdgcn_wmma_f32_16x16x4_f32(false, a, false, b,
                                                  (short)0, c, false, false);
    }
    if (k < kend) {                       // wave-uniform: only the chunk holding the K tail
        const int k0 = k + 2 * half;
        v2f a = {0.0f, 0.0f}, b = {0.0f, 0.0f};
        if (k0 < FC1_K)     { a.x = ap[k];     b.x = bp[k];     }
        if (k0 + 1 < FC1_K) { a.y = ap[k + 1]; b.y = bp[k + 1]; }
        c = __builtin_amdgcn_wmma_f32_16x16x4_f32(false, a, false, b,
                                                  (short)0, c, false, false);
    }

    float* dst = part + (size_t)(kt * FC1_NT + nt) * 256;
    #pragma unroll
    for (int r = 0; r < 8; ++r) {
        const int m = r + 8 * half;       // C/D layout: VGPR r -> rows r, r+8
        dst[m * 16 + lm] = c[r];
    }
}

// ---------------- fc1 deterministic K-reduce + bias + relu ----------------
__global__ __launch_bounds__(256)
void fc1_reduce_kernel(const float* __restrict__ part, const float* __restrict__ bias,
                       float* __restrict__ out) {
    int idx = blockIdx.x * blockDim.x + threadIdx.x;   // 0..2047
    if (idx >= 16 * 128) return;
    int m = idx >> 7, nn = idx & 127;
    if (m >= BATCH || nn >= FC1_N) return;
    int nt = nn >> 4, nl = nn & 15;
    float s = 0.0f;
    for (int kt = 0; kt < FC1_KT; ++kt)
        s += part[(size_t)(kt * FC1_NT + nt) * 256 + m * 16 + nl];
    s += bias[nn];
    out[m * FC1_N + nn] = fmaxf(s, 0.0f);
}

// ---------------- fc2 + fc3 + qnn head (single block, LDS staging) ----------------
__global__ __launch_bounds__(128)
void head_kernel(const float* __restrict__ fc1out,
                 const float* __restrict__ fc2_w, const float* __restrict__ fc2_b,
                 const float* __restrict__ fc3_w, const float* __restrict__ fc3_b,
                 const float* __restrict__ qnn_w1, const float* __restrict__ qnn_w2,
                 float* __restrict__ fs) {
    __shared__ float sh_fc2[BATCH * FC2_N];
    __shared__ float sh_h[BATCH];
    __shared__ float sh_s1[20];
    int tid = threadIdx.x;
    for (int idx = tid; idx < BATCH * FC2_N; idx += 128) {
        int bb = idx / FC2_N, j = idx % FC2_N;
        float s = fc2_b[j];
        for (int t = 0; t < FC1_N; ++t)
            s += fc1out[bb * FC1_N + t] * fc2_w[j * FC1_N + t];
        sh_fc2[idx] = fmaxf(s, 0.0f);
    }
    __syncthreads();
    if (tid < BATCH) {
        float s = fc3_b[0];
        for (int t = 0; t < FC2_N; ++t) s += sh_fc2[tid * FC2_N + t] * fc3_w[t];
        sh_h[tid] = s;
    }
    __syncthreads();
    if (tid < 20) {
        float s = 0.0f;
        for (int bb = 0; bb < BATCH; ++bb) s += qnn_w1[tid * BATCH + bb] * sh_h[bb];
        sh_s1[tid] = tanhf(s);
    }
    __syncthreads();
    if (tid < 5) {
        float s = 0.0f;
        for (int i = 0; i < 20; ++i) s += qnn_w2[tid * 20 + i] * sh_s1[i];
        fs[tid] = tanhf(s);
    }
}

// ---------------- RBF kernel vs train states, per-block partials ----------------
__global__ __launch_bounds__(256)
void rbf_partial_kernel(const float* __restrict__ fs, const float* __restrict__ ts,
                        const float* __restrict__ kw, float* __restrict__ part) {
    __shared__ float red0[256];
    __shared__ float red1[256];
    int tid = threadIdx.x;
    int i = blockIdx.x * 256 + tid;                   // 0..8191 exactly
    float f0 = fs[0], f1 = fs[1], f2 = fs[2], f3 = fs[3], f4 = fs[4];
    const float* t = ts + (size_t)i * 5;
    float d0 = f0 - t[0], d1 = f1 - t[1], d2 = f2 - t[2], d3 = f3 - t[3], d4 = f4 - t[4];
    float dd = d0 * d0 + d1 * d1 + d2 * d2 + d3 * d3 + d4 * d4;
    float Kv = expf(-dd);                             // GAMMA = 1
    red0[tid] = Kv * kw[i];                           // kcls_w row 0
    red1[tid] = Kv * kw[NTRAIN + i];                  // kcls_w row 1
    __syncthreads();
    for (int off = 128; off > 0; off >>= 1) {
        if (tid < off) {
            red0[tid] += red0[tid + off];
            red1[tid] += red1[tid + off];
        }
        __syncthreads();
    }
    if (tid == 0) {
        part[blockIdx.x * 2 + 0] = red0[0];
        part[blockIdx.x * 2 + 1] = red1[0];
    }
}

__global__ __launch_bounds__(32)
void rbf_final_kernel(const float* __restrict__ part, const float* __restrict__ bias,
                      float* __restrict__ out) {
    int c = threadIdx.x;
    if (c < 2) {
        float s = bias[c];
        for (int bk = 0; bk < 32; ++bk) s += part[bk * 2 + c];
        out[c] = s;
    }
}

extern "C" void kernel_launch(void* const* d_in, const int* in_sizes, int n_in,
                              void* d_out, int out_size, void* d_ws, size_t ws_size,
                              hipStream_t stream) {
    const float* x       = (const float*)d_in[0];
    const float* conv1_w = (const float*)d_in[1];
    const float* conv1_b = (const float*)d_in[2];
    const float* conv2_w = (const float*)d_in[3];
    const float* conv2_b = (const float*)d_in[4];
    const float* fc1_w   = (const float*)d_in[5];
    const float* fc1_b   = (const float*)d_in[6];
    const float* fc2_w   = (const float*)d_in[7];
    const float* fc2_b   = (const float*)d_in[8];
    const float* fc3_w   = (const float*)d_in[9];
    const float* fc3_b   = (const float*)d_in[10];
    const float* qnn_w1  = (const float*)d_in[11];
    const float* qnn_w2  = (const float*)d_in[12];
    const float* tstates = (const float*)d_in[13];
    const float* kcls_w  = (const float*)d_in[14];
    const float* kcls_b  = (const float*)d_in[15];

    // workspace layout (floats)
    float* wsf    = (float*)d_ws;
    const size_t n_h1 = (size_t)BATCH * C1_COUT * H1 * W1;   // 922560
    const size_t n_p1 = (size_t)BATCH * C1_COUT * P1H * P1W; // 907740
    const size_t n_h2 = (size_t)BATCH * C2_COUT * H2 * W2;   // 576600
    const size_t n_p2 = (size_t)BATCH * FC1_K;               // 558150
    const size_t n_pt = (size_t)FC1_KT * FC1_NT * 256;       // 262144
    float* h1      = wsf;
    float* p1      = h1 + n_h1;
    float* h2      = p1 + n_p1;
    float* p2      = h2 + n_h2;
    float* fc1part = p2 + n_p2;
    float* fc1out  = fc1part + n_pt;          // 1200
    float* fsbuf   = fc1out + 1200;           // 5 (pad 8)
    float* rbfpart = fsbuf + 8;               // 64

    conv1_kernel<<<(unsigned)((n_h1 + 255) / 256), 256, 0, stream>>>(x, conv1_w, conv1_b, h1);
    maxpool_kernel<<<(unsigned)((n_p1 + 255) / 256), 256, 0, stream>>>(
        h1, p1, C1_COUT, H1, W1, P1H, P1W, (int)n_p1);
    conv2_kernel<<<(unsigned)((n_h2 + 255) / 256), 256, 0, stream>>>(p1, conv2_w, conv2_b, h2);
    maxpool_kernel<<<(unsigned)((n_p2 + 255) / 256), 256, 0, stream>>>(
        h2, p2, C2_COUT, H2, W2, P2H, P2W, (int)n_p2);

    fc1_wmma_kernel<<<dim3(FC1_NT, FC1_KT), 32, 0, stream>>>(p2, fc1_w, fc1part);
    fc1_reduce_kernel<<<8, 256, 0, stream>>>(fc1part, fc1_b, fc1out);

    head_kernel<<<1, 128, 0, stream>>>(fc1out, fc2_w, fc2_b, fc3_w, fc3_b,
                                       qnn_w1, qnn_w2, fsbuf);

    rbf_partial_kernel<<<NTRAIN / 256, 256, 0, stream>>>(fsbuf, tstates, kcls_w, rbfpart);
    rbf_final_kernel<<<1, 32, 0, stream>>>(rbfpart, kcls_b, (float*)d_out);
}